// ProjConvT2I_87746181857682
// MI455X (gfx1250) — compile-verified
//
#include <hip/hip_runtime.h>
#include <hip/hip_bf16.h>
#include <math.h>

// Sizes (fixed by the reference)
#define BI 256
#define BC 256
#define TT 64
#define DD 1024
#define DR 256
#define LL 36
#define KK 3

typedef float v2f __attribute__((ext_vector_type(2)));
typedef float v8f __attribute__((ext_vector_type(8)));

// ---------------------------------------------------------------------------
// Kernel 1: per-caption masked mean, l2norm, q[b] = conv_b . cap_v[b]
// grid = BC blocks, 256 threads; each thread owns 4 channels (strided).
// ---------------------------------------------------------------------------
__global__ void cap_stats_kernel(const float* __restrict__ cap,
                                 const int* __restrict__ lens,
                                 const float* __restrict__ conv_b,
                                 float* __restrict__ cap_full,
                                 float* __restrict__ cap_v,
                                 float* __restrict__ q) {
    const int b = blockIdx.x, tid = threadIdx.x;
    const int len = lens[b];
    const float inv = 1.0f / (float)len;
    const float* base = cap + (size_t)b * TT * DD;

    float acc[4] = {0.f, 0.f, 0.f, 0.f};
    for (int t = 0; t < len; ++t) {
        const float* row = base + t * DD;
#pragma unroll
        for (int j = 0; j < 4; ++j) acc[j] += row[tid + j * 256];
    }
    float cf[4];
    float ss = 0.f;
#pragma unroll
    for (int j = 0; j < 4; ++j) {
        cf[j] = acc[j] * inv;
        ss += cf[j] * cf[j];
        cap_full[(size_t)b * DD + tid + j * 256] = cf[j];
    }

    __shared__ float red[256];
    red[tid] = ss;
    __syncthreads();
    for (int s = 128; s > 0; s >>= 1) {
        if (tid < s) red[tid] += red[tid + s];
        __syncthreads();
    }
    const float invn = 1.0f / (sqrtf(red[0]) + 1e-8f);
    __syncthreads();  // before reusing `red`

    float qq = 0.f;
#pragma unroll
    for (int j = 0; j < 4; ++j) {
        const int d = tid + j * 256;
        const float v = cf[j] * invn;
        cap_v[(size_t)b * DD + d] = v;
        qq += conv_b[d] * v;
    }
    red[tid] = qq;
    __syncthreads();
    for (int s = 128; s > 0; s >>= 1) {
        if (tid < s) red[tid] += red[tid + s];
        __syncthreads();
    }
    if (tid == 0) q[b] = red[0];
}

// ---------------------------------------------------------------------------
// Kernel 2: ||conv_b||^2 (one block)
// ---------------------------------------------------------------------------
__global__ void bb_kernel(const float* __restrict__ conv_b, float* __restrict__ bb) {
    __shared__ float red[256];
    const int tid = threadIdx.x;
    float s = 0.f;
#pragma unroll
    for (int j = 0; j < 4; ++j) {
        const float v = conv_b[tid + j * 256];
        s += v * v;
    }
    red[tid] = s;
    __syncthreads();
    for (int st = 128; st > 0; st >>= 1) {
        if (tid < st) red[tid] += red[tid + st];
        __syncthreads();
    }
    if (tid == 0) *bb = red[0];
}

// ---------------------------------------------------------------------------
// Kernel 3: image pooling. c[i,k,d]: k=0 -> (S - img[35])/L, k=1 -> S/L,
// k=2 -> (S - img[0])/L. grid = BI, 256 threads.
// ---------------------------------------------------------------------------
__global__ void img_pool_kernel(const float* __restrict__ img, float* __restrict__ c) {
    const int i = blockIdx.x, tid = threadIdx.x;
    const float* base = img + (size_t)i * LL * DD;
    const float invL = 1.0f / (float)LL;
#pragma unroll
    for (int j = 0; j < 4; ++j) {
        const int d = tid + j * 256;
        float s = 0.f, v0 = 0.f, v35 = 0.f;
        for (int l = 0; l < LL; ++l) {
            const float x = base[l * DD + d];
            s += x;
            if (l == 0) v0 = x;
            if (l == LL - 1) v35 = x;
        }
        c[((size_t)i * KK + 0) * DD + d] = (s - v35) * invL;
        c[((size_t)i * KK + 1) * DD + d] = s * invL;
        c[((size_t)i * KK + 2) * DD + d] = (s - v0) * invL;
    }
}

// ---------------------------------------------------------------------------
// Generic WMMA f32 GEMM: out[M,N] = A[M,Kd] * B[N,Kd]^T (+ bias[n]).
// One wave per 16x16 tile, 4 waves / block, V_WMMA_F32_16X16X4_F32.
// A frag: lane m = lane&15, K pair = 2*(lane>>4) -> float2 load.
// B frag: symmetric.  C/D: VGPR r -> rows (r, r+8), col = lane&15.
// No divergence anywhere (EXEC must stay all-ones for WMMA).
// ---------------------------------------------------------------------------
__global__ __launch_bounds__(128) void gemm_abt_wmma_kernel(
    const float* __restrict__ A, const float* __restrict__ B,
    const float* __restrict__ bias, float* __restrict__ out,
    int N, int Kd) {
    const int lane = threadIdx.x & 31;
    const int wave = threadIdx.x >> 5;
    const int tile = blockIdx.x * 4 + wave;
    const int ntn  = N >> 4;
    const int tm = tile / ntn;
    const int tn = tile % ntn;

    const int lo  = lane & 15;
    const int hi2 = (lane >> 4) << 1;  // 0 or 2

    const float* ap = A + (size_t)(tm * 16 + lo) * Kd + hi2;
    const float* bp = B + (size_t)(tn * 16 + lo) * Kd + hi2;

    v8f acc = {};
#pragma unroll 8
    for (int k = 0; k < Kd; k += 4) {
        v2f a = *(const v2f*)(ap + k);
        v2f b = *(const v2f*)(bp + k);
        acc = __builtin_amdgcn_wmma_f32_16x16x4_f32(
            /*neg_a=*/false, a, /*neg_b=*/false, b,
            /*c_mod=*/(short)0, acc, /*reuse_a=*/false, /*reuse_b=*/false);
    }

    const int col   = tn * 16 + lo;
    const int rbase = tm * 16 + ((lane >> 4) << 3);
    const float bv = bias ? bias[col] : 0.0f;
#pragma unroll
    for (int r = 0; r < 8; ++r) {
        out[(size_t)(rbase + r) * N + col] = acc[r] + bv;
    }
}

// ---------------------------------------------------------------------------
// Kernel 5: w[b,k] = softmax_k(cap_repr[b,:] . proj_w[k,:] + proj_b[k])
// one block, thread b.
// ---------------------------------------------------------------------------
__global__ void softmax_w_kernel(const float* __restrict__ cap_repr,
                                 const float* __restrict__ proj_w,
                                 const float* __restrict__ proj_b,
                                 float* __restrict__ w) {
    const int b = threadIdx.x;
    float lg[KK];
    const float* cr = cap_repr + (size_t)b * DR;
#pragma unroll
    for (int k = 0; k < KK; ++k) {
        float s = proj_b[k];
        const float* pr = proj_w + k * DR;
        for (int e = 0; e < DR; ++e) s += cr[e] * pr[e];
        lg[k] = s;
    }
    const float mx = fmaxf(lg[0], fmaxf(lg[1], lg[2]));
    const float e0 = expf(lg[0] - mx), e1 = expf(lg[1] - mx), e2 = expf(lg[2] - mx);
    const float inv = 1.0f / (e0 + e1 + e2);
    w[b * KK + 0] = e0 * inv;
    w[b * KK + 1] = e1 * inv;
    w[b * KK + 2] = e2 * inv;
}

// ---------------------------------------------------------------------------
// Kernel 6: per-image 3x3 Gram of cc and Hb = cc . conv_b. grid = BI.
// ---------------------------------------------------------------------------
__global__ void gram_kernel(const float* __restrict__ cc,
                            const float* __restrict__ conv_b,
                            float* __restrict__ G, float* __restrict__ Hb) {
    const int i = blockIdx.x, tid = threadIdx.x;
    const float* c0 = cc + (size_t)i * KK * DD;
    const float* c1 = c0 + DD;
    const float* c2 = c1 + DD;
    float a00 = 0, a01 = 0, a02 = 0, a11 = 0, a12 = 0, a22 = 0;
    float h0 = 0, h1 = 0, h2 = 0;
#pragma unroll
    for (int j = 0; j < 4; ++j) {
        const int e = tid + j * 256;
        const float x0 = c0[e], x1 = c1[e], x2 = c2[e], bv = conv_b[e];
        a00 += x0 * x0; a01 += x0 * x1; a02 += x0 * x2;
        a11 += x1 * x1; a12 += x1 * x2; a22 += x2 * x2;
        h0 += x0 * bv;  h1 += x1 * bv;  h2 += x2 * bv;
    }
    __shared__ float red[9 * 256];
    float vals[9] = {a00, a01, a02, a11, a12, a22, h0, h1, h2};
#pragma unroll
    for (int v = 0; v < 9; ++v) red[v * 256 + tid] = vals[v];
    __syncthreads();
    for (int st = 128; st > 0; st >>= 1) {
        if (tid < st) {
#pragma unroll
            for (int v = 0; v < 9; ++v) red[v * 256 + tid] += red[v * 256 + tid + st];
        }
        __syncthreads();
    }
    if (tid < 6) G[i * 6 + tid] = red[tid * 256];
    if (tid >= 6 && tid < 9) Hb[i * KK + (tid - 6)] = red[tid * 256];
}

// ---------------------------------------------------------------------------
// Kernel 7: sims[i,b] = (sum_k w*P + q[b]) / (sqrt(w'Gw + 2 w.Hb + bb) + eps)
// grid = BI, thread = b.
// ---------------------------------------------------------------------------
__global__ void final_kernel(const float* __restrict__ P, const float* __restrict__ w,
                             const float* __restrict__ q, const float* __restrict__ G,
                             const float* __restrict__ Hb, const float* __restrict__ bbp,
                             float* __restrict__ out) {
    const int i = blockIdx.x, b = threadIdx.x;
    const float w0 = w[b * KK + 0], w1 = w[b * KK + 1], w2 = w[b * KK + 2];
    const float p0 = P[((size_t)i * KK + 0) * BC + b];
    const float p1 = P[((size_t)i * KK + 1) * BC + b];
    const float p2 = P[((size_t)i * KK + 2) * BC + b];
    const float num = w0 * p0 + w1 * p1 + w2 * p2 + q[b];
    const float* g = G + i * 6;
    const float* h = Hb + i * KK;
    const float den2 = w0 * w0 * g[0] + w1 * w1 * g[3] + w2 * w2 * g[5]
                     + 2.0f * (w0 * w1 * g[1] + w0 * w2 * g[2] + w1 * w2 * g[4])
                     + 2.0f * (w0 * h[0] + w1 * h[1] + w2 * h[2]) + *bbp;
    out[(size_t)i * BC + b] = num / (sqrtf(fmaxf(den2, 0.0f)) + 1e-8f);
}

// ---------------------------------------------------------------------------
extern "C" void kernel_launch(void* const* d_in, const int* in_sizes, int n_in,
                              void* d_out, int out_size, void* d_ws, size_t ws_size,
                              hipStream_t stream) {
    const float* img_embed = (const float*)d_in[0];  // (BI, L, D)
    const float* cap_embed = (const float*)d_in[1];  // (BC, T, D)
    const int*   lens      = (const int*)d_in[2];    // (BC,)
    const float* red_w     = (const float*)d_in[3];  // (DR, D)
    const float* red_b     = (const float*)d_in[4];  // (DR,)
    const float* proj_w    = (const float*)d_in[5];  // (K, DR)
    const float* proj_b    = (const float*)d_in[6];  // (K,)
    const float* conv_w    = (const float*)d_in[7];  // (D, D)
    const float* conv_b    = (const float*)d_in[8];  // (D,)
    float* out = (float*)d_out;                      // (BI, BC)

    // Workspace carve-up (floats). Total ~2.37M floats ~= 9.5 MB.
    float* ws = (float*)d_ws;
    float* cap_full = ws;                         // BC*D      = 262144
    float* cap_v    = cap_full + BC * DD;         // BC*D      = 262144
    float* cap_repr = cap_v + BC * DD;            // BC*DR     = 65536
    float* wbuf     = cap_repr + BC * DR;         // BC*K      = 768
    float* cbuf     = wbuf + BC * KK;             // BI*K*D    = 786432
    float* ccbuf    = cbuf + (size_t)BI * KK * DD;// BI*K*D    = 786432
    float* Pbuf     = ccbuf + (size_t)BI * KK * DD; // BI*K*BC = 196608
    float* qbuf     = Pbuf + (size_t)BI * KK * BC;  // BC      = 256
    float* Gbuf     = qbuf + BC;                  // BI*6      = 1536
    float* Hbuf     = Gbuf + BI * 6;              // BI*K      = 768
    float* bbuf     = Hbuf + BI * KK;             // 1

    // 1) caption masked mean + l2norm + q
    cap_stats_kernel<<<BC, 256, 0, stream>>>(cap_embed, lens, conv_b,
                                             cap_full, cap_v, qbuf);
    // 2) ||conv_b||^2
    bb_kernel<<<1, 256, 0, stream>>>(conv_b, bbuf);
    // 3) cap_repr = cap_full @ red_w^T + red_b  (M=256,N=256,Kd=1024)
    gemm_abt_wmma_kernel<<<(BC / 16) * (DR / 16) / 4, 128, 0, stream>>>(
        cap_full, red_w, red_b, cap_repr, DR, DD);
    // 4) w = softmax(cap_repr @ proj_w^T + proj_b)
    softmax_w_kernel<<<1, BC, 0, stream>>>(cap_repr, proj_w, proj_b, wbuf);
    // 5) image shifted-sum pooling -> c
    img_pool_kernel<<<BI, 256, 0, stream>>>(img_embed, cbuf);
    // 6) cc = c @ conv_w^T  (M=768,N=1024,Kd=1024)
    gemm_abt_wmma_kernel<<<(BI * KK / 16) * (DD / 16) / 4, 128, 0, stream>>>(
        cbuf, conv_w, nullptr, ccbuf, DD, DD);
    // 7) per-image Gram + Hb
    gram_kernel<<<BI, 256, 0, stream>>>(ccbuf, conv_b, Gbuf, Hbuf);
    // 8) P = cc @ cap_v^T  (M=768,N=256,Kd=1024)
    gemm_abt_wmma_kernel<<<(BI * KK / 16) * (BC / 16) / 4, 128, 0, stream>>>(
        ccbuf, cap_v, nullptr, Pbuf, BC, DD);
    // 9) final combine
    final_kernel<<<BI, BC, 0, stream>>>(Pbuf, wbuf, qbuf, Gbuf, Hbuf, bbuf, out);

    (void)in_sizes; (void)n_in; (void)out_size; (void)ws_size;
}